// TrainableTree_33784212750822
// MI455X (gfx1250) — compile-verified
//
#include <hip/hip_runtime.h>
#include <hip/hip_bf16.h>
#include <math.h>

// ---------------------------------------------------------------------------
// Problem constants (match the JAX reference)
// ---------------------------------------------------------------------------
static constexpr int Bn = 1024;   // batch
static constexpr int Dn = 512;    // feature dim
static constexpr int Kn = 255;    // internal tree nodes
static constexpr int Ln = 256;    // leaves
static constexpr int Cn = 80;     // output channels
static constexpr int DEPTH = 8;

static constexpr int XS_STRIDE = 516;      // LDS row stride (floats), bank-conflict free
static constexpr int NBTILE = Bn / 16;     // 64 batch tiles
static constexpr int LGROUPS = 8;          // leaf groups in the fused GEMM
static constexpr int LPG = Ln / LGROUPS;   // 32 leaves per group

typedef __attribute__((ext_vector_type(2))) float v2f;
typedef __attribute__((ext_vector_type(8))) float v8f;

// ws layout (in floats)
static constexpr size_t WS_INVNORM = 0;                       // 256
static constexpr size_t WS_PROB    = 256;                     // 1024*256
static constexpr size_t WS_PART    = 256 + (size_t)Bn * Ln;   // LGROUPS*1024*80

__device__ __forceinline__ float fast_sigmoid(float v) {
  // jax.nn.sigmoid equivalent; v_exp_f32 + single v_rcp_f32 (no IEEE div fixup)
  return __builtin_amdgcn_rcpf(1.0f + __expf(-v));
}

// ---------------------------------------------------------------------------
// Kernel A: inv_norm[k] = 1 / ||node_embeddings[k]||
// ---------------------------------------------------------------------------
__global__ void tt_invnorm_kernel(const float* __restrict__ ne,
                                  float* __restrict__ inv_norm) {
  __shared__ float red[256];
  const int k = blockIdx.x;           // 0..254
  const int t = threadIdx.x;          // 0..255
  const float* row = ne + (size_t)k * Dn;
  float s = 0.f;
  for (int d = t; d < Dn; d += 256) {
    float v = row[d];
    s += v * v;
  }
  red[t] = s;
  __syncthreads();
  for (int off = 128; off > 0; off >>= 1) {
    if (t < off) red[t] += red[t + off];
    __syncthreads();
  }
  if (t == 0) inv_norm[k] = rsqrtf(red[0]);
}

// ---------------------------------------------------------------------------
// Kernel B: node_sim[b,k] = sigmoid(inv_norm[k] * dot(x[b], ne[k]))
//   One block per 16-row batch tile; 8 waves, 2 k-tiles (16 cols) per wave.
// ---------------------------------------------------------------------------
__global__ void tt_nodesim_kernel(const float* __restrict__ x,
                                  const float* __restrict__ ne,
                                  const float* __restrict__ inv_norm,
                                  float* __restrict__ node_sim /* B x 255 */) {
  __shared__ float xs[16 * XS_STRIDE];
  const int btile = blockIdx.x;                 // 0..63
  const int tid = threadIdx.x;                  // 0..255
  // stage X tile (16 x 512) into LDS, coalesced
  for (int i = tid; i < 16 * Dn; i += 256) {
    const int r = i >> 9;
    const int d = i & (Dn - 1);
    xs[r * XS_STRIDE + d] = x[(size_t)(btile * 16 + r) * Dn + d];
  }
  __syncthreads();

  const int wave = tid >> 5;                    // 0..7
  const int lane = tid & 31;
  const int m  = lane & 15;                     // row within tile / column within tile
  const int kh = lane >> 4;                     // K-half select (0/1)
  const float* xrow = xs + m * XS_STRIDE + 2 * kh;

  for (int t = 0; t < 2; ++t) {
    const int kt  = wave * 2 + t;               // k-tile 0..15
    const int col = kt * 16 + m;                // node index 0..255 (255 is padding)
    const int nrow = col < Kn ? col : (Kn - 1); // clamp OOB row (result masked out)
    const float* nerow = ne + (size_t)nrow * Dn + 2 * kh;

    v8f acc = {};
    #pragma unroll 4
    for (int kb = 0; kb < Dn; kb += 4) {
      v2f a, b;
      a[0] = xrow[kb];  a[1] = xrow[kb + 1];
      b[0] = nerow[kb]; b[1] = nerow[kb + 1];
      acc = __builtin_amdgcn_wmma_f32_16x16x4_f32(
          /*neg_a=*/false, a, /*neg_b=*/false, b,
          /*c_mod=*/(short)0, acc, /*reuse_a=*/false, /*reuse_b=*/false);
    }
    if (col < Kn) {
      const float inv = inv_norm[col];
      #pragma unroll
      for (int r = 0; r < 8; ++r) {
        const int brow = btile * 16 + r + 8 * kh;
        node_sim[(size_t)brow * Kn + col] = fast_sigmoid(acc[r] * inv);
      }
    }
  }
}

// ---------------------------------------------------------------------------
// Kernel C: prob[b, leaf] = product of 8 path factors (binary-tree walk).
//   One block per batch row; ns row cached in LDS.
// ---------------------------------------------------------------------------
__global__ void tt_prob_kernel(const float* __restrict__ node_sim,
                               float* __restrict__ prob /* B x 256 */) {
  __shared__ float ns[256];
  const int b = blockIdx.x;
  const int t = threadIdx.x;                    // 0..255 == leaf id
  if (t < Kn) ns[t] = node_sim[(size_t)b * Kn + t];
  __syncthreads();

  int node = 0;
  float p = 1.0f;
  #pragma unroll
  for (int d = 0; d < DEPTH; ++d) {
    const int bit = (t >> (DEPTH - 1 - d)) & 1;
    const float s = ns[node];
    p *= bit ? (1.0f - s) : s;
    node = 2 * node + 1 + bit;
  }
  prob[(size_t)b * Ln + t] = p;
}

// ---------------------------------------------------------------------------
// Kernel D: fused leaf GEMM + prob-weighted reduction.
//   partial[g][b][c] = sum_{leaf in group g} prob[b,leaf] *
//                      (dot(x[b], W[leaf*C+c]) + bias[leaf*C+c])
//   grid = (64 btiles, 8 leaf groups); block = 160 (5 waves, one per n-tile).
//   Two independent WMMA accumulation chains (even/odd k-steps) to cover
//   matrix-pipe latency when few waves are resident per SIMD.
// ---------------------------------------------------------------------------
__global__ void __launch_bounds__(160)
tt_fused_leaf_gemm(const float* __restrict__ x,
                   const float* __restrict__ W,
                   const float* __restrict__ bias,
                   const float* __restrict__ prob,
                   float* __restrict__ partials) {
  __shared__ float xs[16 * XS_STRIDE];   // 16 x 512 X tile (padded)
  __shared__ float ps[16 * LPG];         // prob tile: 16 rows x 32 leaves

  const int btile = blockIdx.x;          // 0..63
  const int lg    = blockIdx.y;          // 0..7
  const int tid   = threadIdx.x;         // 0..159

  for (int i = tid; i < 16 * Dn; i += 160) {
    const int r = i >> 9;
    const int d = i & (Dn - 1);
    xs[r * XS_STRIDE + d] = x[(size_t)(btile * 16 + r) * Dn + d];
  }
  for (int i = tid; i < 16 * LPG; i += 160) {
    const int r  = i / LPG;
    const int lf = i % LPG;
    ps[r * LPG + lf] = prob[(size_t)(btile * 16 + r) * Ln + lg * LPG + lf];
  }
  __syncthreads();

  const int wave = tid >> 5;             // n-tile 0..4 (channels wave*16..+15)
  const int lane = tid & 31;
  const int m  = lane & 15;
  const int kh = lane >> 4;
  const float* xrow = xs + m * XS_STRIDE + 2 * kh;

  v8f acc = {};                          // prob-weighted running sum (16x16 tile)

  for (int lf = 0; lf < LPG; ++lf) {
    const int leaf = lg * LPG + lf;
    const int o    = leaf * Cn + wave * 16 + m;       // output row of leaf_W
    const float* wrow = W + (size_t)o * Dn + 2 * kh;

    v8f d0 = {}, d1 = {};                // two independent accumulation chains
    #pragma unroll 2
    for (int kb = 0; kb < Dn; kb += 8) {
      v2f a0, b0, a1, b1;
      a0[0] = xrow[kb];     a0[1] = xrow[kb + 1];
      b0[0] = wrow[kb];     b0[1] = wrow[kb + 1];
      a1[0] = xrow[kb + 4]; a1[1] = xrow[kb + 5];
      b1[0] = wrow[kb + 4]; b1[1] = wrow[kb + 5];
      d0 = __builtin_amdgcn_wmma_f32_16x16x4_f32(
          false, a0, false, b0, (short)0, d0, false, false);
      d1 = __builtin_amdgcn_wmma_f32_16x16x4_f32(
          false, a1, false, b1, (short)0, d1, false, false);
    }

    const float bv = bias[o];            // bias depends only on column c
    #pragma unroll
    for (int r = 0; r < 8; ++r) {
      const float p = ps[(r + 8 * kh) * LPG + lf];
      acc[r] += p * ((d0[r] + d1[r]) + bv);
    }
  }

  float* outp = partials + (size_t)lg * Bn * Cn;
  #pragma unroll
  for (int r = 0; r < 8; ++r) {
    const int brow = btile * 16 + r + 8 * kh;
    outp[(size_t)brow * Cn + wave * 16 + m] = acc[r];
  }
}

// ---------------------------------------------------------------------------
// Kernel E: out1[b,c] = sum over 8 leaf-group partials (deterministic reduce)
// ---------------------------------------------------------------------------
__global__ void tt_reduce_kernel(const float* __restrict__ partials,
                                 float* __restrict__ out1) {
  const int idx = blockIdx.x * 256 + threadIdx.x;   // 0..81919
  if (idx >= Bn * Cn) return;
  float s = 0.f;
  #pragma unroll
  for (int g = 0; g < LGROUPS; ++g) s += partials[(size_t)g * Bn * Cn + idx];
  out1[idx] = s;
}

// ---------------------------------------------------------------------------
extern "C" void kernel_launch(void* const* d_in, const int* in_sizes, int n_in,
                              void* d_out, int out_size, void* d_ws, size_t ws_size,
                              hipStream_t stream) {
  const float* x      = (const float*)d_in[0];   // B x D
  const float* ne     = (const float*)d_in[1];   // K x D
  const float* leaf_W = (const float*)d_in[2];   // (L*C) x D
  const float* leaf_b = (const float*)d_in[3];   // L*C
  // d_in[4] = res_path (L x K int32) -- recomputed analytically on device.

  float* out1     = (float*)d_out;               // B x C
  float* node_sim = (float*)d_out + (size_t)Bn * Cn;  // B x K

  float* ws       = (float*)d_ws;
  float* inv_norm = ws + WS_INVNORM;             // 256
  float* prob     = ws + WS_PROB;                // B x L
  float* partials = ws + WS_PART;                // LGROUPS x B x C

  tt_invnorm_kernel<<<Kn, 256, 0, stream>>>(ne, inv_norm);
  tt_nodesim_kernel<<<NBTILE, 256, 0, stream>>>(x, ne, inv_norm, node_sim);
  tt_prob_kernel<<<Bn, 256, 0, stream>>>(node_sim, prob);
  tt_fused_leaf_gemm<<<dim3(NBTILE, LGROUPS), 160, 0, stream>>>(
      x, leaf_W, leaf_b, prob, partials);
  tt_reduce_kernel<<<(Bn * Cn + 255) / 256, 256, 0, stream>>>(partials, out1);
}